// FilterDetections_9509057593763
// MI455X (gfx1250) — compile-verified
//
#include <hip/hip_runtime.h>
#include <stdint.h>

// FilterDetections (per-class sequential NMS + global top-k) for MI455X/gfx1250.
// No matmul structure -> WMMA not applicable; CDNA5-specific paths used instead:
//   * global_load_async_to_lds_b128 + s_wait_asynccnt (async box staging to LDS)
//   * wave32 shuffle reductions (warp width 32, not 64)

namespace {
constexpr int   kB       = 2;
constexpr int   kN       = 50000;
constexpr int   kC       = 15;
constexpr int   kMaxDet  = 300;
constexpr float kScoreThr = 0.05f;
constexpr float kIouThr   = 0.5f;
constexpr float kNeg      = -1.0e9f;

constexpr int kTpb1   = 1024;                       // 32 wave32 waves = full WGP
constexpr int kItems  = (kN + kTpb1 - 1) / kTpb1;   // 49 scores in registers/thread
constexpr int kCacheN = 3072;                       // boxes staged in LDS (48 KB)

constexpr int kTpb2 = 512;
constexpr int kCand = kC * kMaxDet;                 // 4500 candidates per image
}  // namespace

__device__ __forceinline__ void takeBetter(float& bv, int& bi, float v, int i) {
  // value-descending, index-ascending tie break (matches jnp.argmax / stable top_k)
  if (v > bv || (v == bv && i < bi)) { bv = v; bi = i; }
}

// ---------------------------------------------------------------------------
// Phase 1: one workgroup per (batch, class). Sequential NMS, 300 picks.
// ---------------------------------------------------------------------------
__global__ __launch_bounds__(kTpb1) void nms_per_class(
    const float4* __restrict__ boxes,   // [B,N] float4 (y1,x1,y2,x2)
    const float*  __restrict__ cls,     // [B,N,C]
    float* __restrict__ candS,          // [B*C, 300]
    int*   __restrict__ candI)          // [B*C, 300]
{
  __shared__ float4 sBox[kCacheN];
  __shared__ float  sRv[kTpb1 / 32];
  __shared__ int    sRi[kTpb1 / 32];
  __shared__ float  sPickS;
  __shared__ int    sPickI;
  __shared__ float4 sPickB;

  const int bc   = blockIdx.x;
  const int b    = bc / kC;
  const int c    = bc % kC;
  const int tid  = threadIdx.x;
  const int lane = tid & 31;
  const int wave = tid >> 5;

  const float4* gbox = boxes + (size_t)b * kN;
  const float*  gcls = cls + ((size_t)b * kN) * kC + c;

  // --- Stage first kCacheN boxes into LDS with CDNA5 async global->LDS copies.
  for (int i = tid; i < kCacheN; i += kTpb1) {
    unsigned      ldsAddr = (unsigned)(uintptr_t)(&sBox[i]);  // wave-relative LDS byte addr
    const float4* src     = gbox + i;
    asm volatile("global_load_async_to_lds_b128 %0, %1, off"
                 :
                 : "v"(ldsAddr), "v"(src)
                 : "memory");
  }
#if __has_builtin(__builtin_amdgcn_s_wait_asynccnt)
  __builtin_amdgcn_s_wait_asynccnt(0);
#else
  asm volatile("s_wait_asynccnt 0" ::: "memory");
#endif
  __syncthreads();

  // --- Thresholded scores resident in registers (statically unrolled array).
  float s[kItems];
#pragma unroll
  for (int k = 0; k < kItems; ++k) {
    const int   j = tid + k * kTpb1;
    const float v = (j < kN) ? gcls[(size_t)j * kC] : kNeg;
    s[k] = (v > kScoreThr) ? v : kNeg;
  }

  float* myS = candS + (size_t)bc * kMaxDet;
  int*   myI = candI + (size_t)bc * kMaxDet;

  for (int it = 0; it < kMaxDet; ++it) {
    // ---- workgroup argmax (local regs -> wave32 shuffle -> LDS -> wave0) ----
    float bv = -3.4e38f;
    int   bi = 0x7fffffff;
#pragma unroll
    for (int k = 0; k < kItems; ++k) takeBetter(bv, bi, s[k], tid + k * kTpb1);

#pragma unroll
    for (int off = 16; off > 0; off >>= 1) {
      const float ov = __shfl_xor(bv, off, 32);
      const int   oi = __shfl_xor(bi, off, 32);
      takeBetter(bv, bi, ov, oi);
    }
    if (lane == 0) { sRv[wave] = bv; sRi[wave] = bi; }
    __syncthreads();
    if (wave == 0) {
      float v  = sRv[lane];   // exactly 32 waves -> every lane has an entry
      int   i2 = sRi[lane];
#pragma unroll
      for (int off = 16; off > 0; off >>= 1) {
        const float ov = __shfl_xor(v, off, 32);
        const int   oi = __shfl_xor(i2, off, 32);
        takeBetter(v, i2, ov, oi);
      }
      if (lane == 0) {
        sPickS = v;
        sPickI = i2;
        sPickB = (i2 < kCacheN) ? sBox[i2] : gbox[i2];
      }
    }
    __syncthreads();
    const float  ps = sPickS;
    const int    pi = sPickI;
    const float4 pb = sPickB;

    if (ps <= kNeg * 0.5f) {
      // Nothing valid remains: reference keeps picking (idx 0, NEG). Fill & stop.
      for (int t2 = it + tid; t2 < kMaxDet; t2 += kTpb1) { myS[t2] = kNeg; myI[t2] = 0; }
      break;  // uniform across the block (decision from shared value)
    }
    if (tid == 0) { myS[it] = ps; myI[it] = pi; }  // score recorded pre-suppression

    // ---- IoU suppression against picked box ----
    const float pa = (pb.z - pb.x) * (pb.w - pb.y);
#pragma unroll
    for (int k = 0; k < kItems; ++k) {
      const int j = tid + k * kTpb1;
      if (j == pi) { s[k] = kNeg; continue; }
      if (j < kN && s[k] > kNeg * 0.5f) {
        const float4 ob   = (j < kCacheN) ? sBox[j] : gbox[j];  // LDS hit or L2-resident
        const float  y1   = fmaxf(pb.x, ob.x);
        const float  x1   = fmaxf(pb.y, ob.y);
        const float  y2   = fminf(pb.z, ob.z);
        const float  x2   = fminf(pb.w, ob.w);
        const float  intr = fmaxf(y2 - y1, 0.0f) * fmaxf(x2 - x1, 0.0f);
        const float  oa   = (ob.z - ob.x) * (ob.w - ob.y);
        const float  iou  = intr / (pa + oa - intr + 1e-9f);
        if (iou > kIouThr) s[k] = kNeg;
      }
    }
  }
}

// ---------------------------------------------------------------------------
// Phase 2: one workgroup per image. Stable top-300 of 4500 candidates + gather.
// ---------------------------------------------------------------------------
__global__ __launch_bounds__(kTpb2) void topk_gather(
    const float4* __restrict__ boxes,
    const float*  __restrict__ candS,
    const int*    __restrict__ candI,
    float* __restrict__ out)
{
  __shared__ float cs[kCand];
  __shared__ float sRv[kTpb2 / 32];
  __shared__ int   sRi[kTpb2 / 32];
  __shared__ float sBs;
  __shared__ int   sBk;
  constexpr int nWaves = kTpb2 / 32;  // 16

  const int b    = blockIdx.x;
  const int tid  = threadIdx.x;
  const int lane = tid & 31;
  const int wave = tid >> 5;

  const float*  myS  = candS + (size_t)b * kCand;  // contiguous: [(b*C+c)*300 + it]
  const int*    myI  = candI + (size_t)b * kCand;
  const float4* gbox = boxes + (size_t)b * kN;

  for (int k = tid; k < kCand; k += kTpb2) cs[k] = myS[k];
  __syncthreads();

  // Output layout: boxes[2,300,4] | scores[2,300] | labels[2,300], flat floats.
  float* oB = out + (size_t)b * kMaxDet * 4;
  float* oS = out + (size_t)kB * kMaxDet * 4 + (size_t)b * kMaxDet;
  float* oL = out + (size_t)kB * kMaxDet * 5 + (size_t)b * kMaxDet;

  for (int t = 0; t < kMaxDet; ++t) {
    float bv = -3.4e38f;
    int   bk = 0x7fffffff;
    for (int k = tid; k < kCand; k += kTpb2) takeBetter(bv, bk, cs[k], k);
#pragma unroll
    for (int off = 16; off > 0; off >>= 1) {
      const float ov = __shfl_xor(bv, off, 32);
      const int   oi = __shfl_xor(bk, off, 32);
      takeBetter(bv, bk, ov, oi);
    }
    if (lane == 0) { sRv[wave] = bv; sRi[wave] = bk; }
    __syncthreads();
    if (wave == 0) {
      float v  = (lane < nWaves) ? sRv[lane] : -3.4e38f;
      int   i2 = (lane < nWaves) ? sRi[lane] : 0x7fffffff;
#pragma unroll
      for (int off = 16; off > 0; off >>= 1) {
        const float ov = __shfl_xor(v, off, 32);
        const int   oi = __shfl_xor(i2, off, 32);
        takeBetter(v, i2, ov, oi);
      }
      if (lane == 0) { sBs = v; sBk = i2; }
    }
    __syncthreads();
    if (tid == 0) {
      const float bs = sBs;
      const int   k  = sBk;
      cs[k] = -3.4e38f;  // strictly below kNeg -> never re-picked
      if (bs > kNeg * 0.5f) {
        const int    anchor = myI[k];
        const float4 bx     = gbox[anchor];
        oB[(size_t)t * 4 + 0] = bx.x;
        oB[(size_t)t * 4 + 1] = bx.y;
        oB[(size_t)t * 4 + 2] = bx.z;
        oB[(size_t)t * 4 + 3] = bx.w;
        oS[t] = bs;
        oL[t] = (float)(k / kMaxDet);
      } else {
        oB[(size_t)t * 4 + 0] = -1.0f;
        oB[(size_t)t * 4 + 1] = -1.0f;
        oB[(size_t)t * 4 + 2] = -1.0f;
        oB[(size_t)t * 4 + 3] = -1.0f;
        oS[t] = -1.0f;
        oL[t] = -1.0f;
      }
    }
    __syncthreads();
  }
}

extern "C" void kernel_launch(void* const* d_in, const int* in_sizes, int n_in,
                              void* d_out, int out_size, void* d_ws, size_t ws_size,
                              hipStream_t stream) {
  (void)in_sizes; (void)n_in; (void)out_size; (void)ws_size;
  const float4* boxes = (const float4*)d_in[0];   // [B,N,4] fp32
  const float*  cls   = (const float*)d_in[1];    // [B,N,C] fp32

  float* candS = (float*)d_ws;                                       // 9000 floats
  int*   candI = (int*)((char*)d_ws + sizeof(float) * (size_t)kB * kC * kMaxDet);

  nms_per_class<<<kB * kC, kTpb1, 0, stream>>>(boxes, cls, candS, candI);
  topk_gather<<<kB, kTpb2, 0, stream>>>(boxes, candS, candI, (float*)d_out);
}